// GraphEncoder_66125316489695
// MI455X (gfx1250) — compile-verified
//
#include <hip/hip_runtime.h>
#include <hip/hip_bf16.h>

typedef __attribute__((ext_vector_type(2))) float v2f;
typedef __attribute__((ext_vector_type(8))) float v8f;

#define NN 50000
#define NE 800000
#define IN_DIM 128
#define HID_DIM 64
#define LAT_DIM 32

// ---------------- degree / normalization ----------------
__global__ void k_init_deg(float* deg, int n) {
  int i = blockIdx.x * blockDim.x + threadIdx.x;
  if (i < n) deg[i] = 1.0f;  // self-loop weight
}

__global__ void k_accum_deg(float* deg, const int* __restrict__ col,
                            const float* __restrict__ ew, int e) {
  int i = blockIdx.x * blockDim.x + threadIdx.x;
  if (i < e) atomicAdd(&deg[col[i]], ew[i]);
}

__global__ void k_deg_to_dinv(float* deg, int n) {
  int i = blockIdx.x * blockDim.x + threadIdx.x;
  if (i < n) {
    float d = deg[i];
    deg[i] = (d > 0.0f) ? rsqrtf(d) : 0.0f;
  }
}

// ---------------- fp32 WMMA GEMM: out[M,N] = X[M,K] @ W[K,N] ----------------
// V_WMMA_F32_16X16X4_F32; W staged in LDS once per block, PRE-PAIRED along K:
// sW[(k/2)*N + n] = {W[k][n], W[k+1][n]}  (ka is always even), so each B
// operand is one aligned ds_load_b64 straight into an even VGPR pair.
// k-loop fully unrolled; A loaded once per k-step (global_load_b64) and
// reused across all N/16 tiles. Block = 160 threads = 5 waves, each wave
// owns one 16-row tile (625 * 5 * 16 = 50000 exactly -> EXEC all-1s at WMMA).
// A layout (ISA 7.12.2): lanes 0-15 carry K=k,k+1; lanes 16-31 K=k+2,k+3.
// C/D layout: VGPR v holds M=v (lanes 0-15) / M=8+v (lanes 16-31), N=lane%16.
template <int K, int N>
__global__ __launch_bounds__(160) void k_gemm_wmma_f32(
    const float* __restrict__ X, const float* __restrict__ Wg,
    float* __restrict__ out) {
  __shared__ v2f sW[(K / 2) * N];
  const int tid = threadIdx.x;
  for (int i = tid; i < (K / 2) * N; i += 160) {
    const int kk = (i / N) * 2;  // even k
    const int n = i % N;
    v2f p;
    p.x = Wg[kk * N + n];
    p.y = Wg[(kk + 1) * N + n];
    sW[i] = p;
  }
  __syncthreads();

  const int lane = tid & 31;
  const int ml = lane & 15;
  const bool hi = lane >= 16;
  const int wave = tid >> 5;  // 0..4
  const int m0 = (blockIdx.x * 5 + wave) * 16;
  const float* xrow = X + (size_t)(m0 + ml) * K;

  constexpr int NT = N / 16;
  v8f acc[NT];
#pragma unroll
  for (int t = 0; t < NT; ++t) acc[t] = (v8f){};

#pragma unroll
  for (int k = 0; k < K; k += 4) {
    const int ka = k + (hi ? 2 : 0);  // even
    v2f a;
    a.x = xrow[ka];      // contiguous pair -> global_load_b64
    a.y = xrow[ka + 1];
    const int brow = (ka >> 1) * N;
#pragma unroll
    for (int t = 0; t < NT; ++t) {
      v2f b = sW[brow + t * 16 + ml];  // one ds_load_b64, even-aligned pair
      acc[t] = __builtin_amdgcn_wmma_f32_16x16x4_f32(
          /*neg_a=*/false, a, /*neg_b=*/false, b,
          /*c_mod=*/(short)0, acc[t], /*reuse_a=*/false, /*reuse_b=*/false);
    }
  }

  const int mb = m0 + (hi ? 8 : 0);
#pragma unroll
  for (int t = 0; t < NT; ++t) {
    const int c = t * 16 + ml;
#pragma unroll
    for (int v = 0; v < 8; ++v) out[(size_t)(mb + v) * N + c] = acc[t][v];
  }
}

// ---------------- aggregation (layer 1, D = 64) ----------------
// self-loop init, vectorized float4: 16 float4 per node
__global__ void k_init_agg64_v4(float4* __restrict__ agg, const float4* __restrict__ Xt,
                                const float* __restrict__ dinv, int n4) {
  int i = blockIdx.x * blockDim.x + threadIdx.x;
  if (i < n4) {
    float s = dinv[i >> 4];
    float ss = s * s;
    float4 x = Xt[i];
    agg[i] = make_float4(x.x * ss, x.y * ss, x.z * ss, x.w * ss);
  }
}

__global__ void k_edge_agg64(float* __restrict__ agg, const float* __restrict__ Xt,
                             const int* __restrict__ row, const int* __restrict__ col,
                             const float* __restrict__ ew, const float* __restrict__ dinv,
                             int E) {
  int e = blockIdx.x * (blockDim.x >> 5) + (threadIdx.x >> 5);  // one wave per edge
  if (e >= E) return;
  int lane = threadIdx.x & 31;
  int r = row[e], c = col[e];
  float coef = dinv[r] * ew[e] * dinv[c];
  const float* src = Xt + (size_t)r * 64;
  float* dst = agg + (size_t)c * 64;
  atomicAdd(&dst[lane], src[lane] * coef);
  atomicAdd(&dst[lane + 32], src[lane + 32] * coef);
}

// bias + ReLU, vectorized: 16 float4 per node
__global__ void k_bias_relu64_v4(float4* a, const float4* __restrict__ b, int n4) {
  int i = blockIdx.x * blockDim.x + threadIdx.x;
  if (i < n4) {
    float4 v = a[i];
    float4 bb = b[i & 15];
    a[i] = make_float4(fmaxf(v.x + bb.x, 0.f), fmaxf(v.y + bb.y, 0.f),
                       fmaxf(v.z + bb.z, 0.f), fmaxf(v.w + bb.w, 0.f));
  }
}

// ---------------- aggregation (layers 2a/2b fused, D = 32) ----------------
// bias + self-loop init, vectorized: 8 float4 per node, both outputs
__global__ void k_init_out2_v4(float4* __restrict__ mu, float4* __restrict__ lv,
                               const float4* __restrict__ Xmu, const float4* __restrict__ Xlv,
                               const float4* __restrict__ bmu, const float4* __restrict__ blv,
                               const float* __restrict__ dinv, int n4) {
  int i = blockIdx.x * blockDim.x + threadIdx.x;
  if (i < n4) {
    float s = dinv[i >> 3];
    float ss = s * s;
    int d = i & 7;
    float4 xm = Xmu[i], xl = Xlv[i], bm = bmu[d], bl = blv[d];
    mu[i] = make_float4(bm.x + xm.x * ss, bm.y + xm.y * ss,
                        bm.z + xm.z * ss, bm.w + xm.w * ss);
    lv[i] = make_float4(bl.x + xl.x * ss, bl.y + xl.y * ss,
                        bl.z + xl.z * ss, bl.w + xl.w * ss);
  }
}

__global__ void k_edge_agg_lat(float* __restrict__ mu, float* __restrict__ lv,
                               const float* __restrict__ Xmu, const float* __restrict__ Xlv,
                               const int* __restrict__ row, const int* __restrict__ col,
                               const float* __restrict__ ew, const float* __restrict__ dinv,
                               int E) {
  int e = blockIdx.x * (blockDim.x >> 5) + (threadIdx.x >> 5);  // one wave per edge
  if (e >= E) return;
  int lane = threadIdx.x & 31;
  int r = row[e], c = col[e];
  float coef = dinv[r] * ew[e] * dinv[c];
  size_t so = (size_t)r * 32 + lane;
  size_t doff = (size_t)c * 32 + lane;
  atomicAdd(&mu[doff], Xmu[so] * coef);
  atomicAdd(&lv[doff], Xlv[so] * coef);
}

// ---------------- launch ----------------
extern "C" void kernel_launch(void* const* d_in, const int* in_sizes, int n_in,
                              void* d_out, int out_size, void* d_ws, size_t ws_size,
                              hipStream_t stream) {
  (void)in_sizes; (void)n_in; (void)out_size; (void)ws_size;

  const float* Y    = (const float*)d_in[0];   // [50000,128]
  const int*   eidx = (const int*)d_in[1];     // [2,800000]
  const float* ew   = (const float*)d_in[2];   // [800000]
  const float* W1   = (const float*)d_in[3];   // [128,64]
  const float* b1   = (const float*)d_in[4];   // [64]
  const float* Wmu  = (const float*)d_in[5];   // [64,32]
  const float* bmu  = (const float*)d_in[6];   // [32]
  const float* Wlv  = (const float*)d_in[7];   // [64,32]
  const float* blv  = (const float*)d_in[8];   // [32]

  const int* row = eidx;
  const int* col = eidx + NE;

  // workspace layout (floats); all offsets 16-byte aligned
  float* ws    = (float*)d_ws;
  float* dinv  = ws;                     // 50000 (padded to 50048)
  float* Xt1   = ws + 50048;             // 50000*64 (reused for Xmu/Xlv)
  float* agg   = Xt1 + (size_t)NN * 64;  // 50000*64 (becomes H in place)
  float* Xmu   = Xt1;                    // 50000*32
  float* Xlv   = Xt1 + (size_t)NN * 32;  // 50000*32

  float* mu_out = (float*)d_out;                    // [50000,32]
  float* lv_out = (float*)d_out + (size_t)NN * 32;  // [50000,32]

  const int TB = 256;
  const int gN    = (NN + TB - 1) / TB;
  const int gE    = (NE + TB - 1) / TB;
  const int gE_w  = (NE + (TB / 32) - 1) / (TB / 32);  // one wave per edge
  const int n4_64 = NN * 16;                           // float4 count, D=64
  const int n4_32 = NN * 8;                            // float4 count, D=32
  const int g64v  = (n4_64 + TB - 1) / TB;
  const int g32v  = (n4_32 + TB - 1) / TB;
  const int gGemm = NN / 80;  // 625 blocks * 5 waves * 16 rows = 50000

  // 1) symmetric normalization coefficients
  k_init_deg<<<gN, TB, 0, stream>>>(dinv, NN);
  k_accum_deg<<<gE, TB, 0, stream>>>(dinv, col, ew, NE);
  k_deg_to_dinv<<<gN, TB, 0, stream>>>(dinv, NN);

  // 2) layer 1: transform (WMMA f32 + LDS-staged, K-paired W), aggregate, bias+ReLU
  k_gemm_wmma_f32<IN_DIM, HID_DIM><<<gGemm, 160, 0, stream>>>(Y, W1, Xt1);
  k_init_agg64_v4<<<g64v, TB, 0, stream>>>((float4*)agg, (const float4*)Xt1, dinv, n4_64);
  k_edge_agg64<<<gE_w, TB, 0, stream>>>(agg, Xt1, row, col, ew, dinv, NE);
  k_bias_relu64_v4<<<g64v, TB, 0, stream>>>((float4*)agg, (const float4*)b1, n4_64);

  // 3) layer 2: two WMMA transforms, fused mu/logvar aggregation into d_out
  k_gemm_wmma_f32<HID_DIM, LAT_DIM><<<gGemm, 160, 0, stream>>>(agg, Wmu, Xmu);
  k_gemm_wmma_f32<HID_DIM, LAT_DIM><<<gGemm, 160, 0, stream>>>(agg, Wlv, Xlv);
  k_init_out2_v4<<<g32v, TB, 0, stream>>>((float4*)mu_out, (float4*)lv_out,
                                          (const float4*)Xmu, (const float4*)Xlv,
                                          (const float4*)bmu, (const float4*)blv,
                                          dinv, n4_32);
  k_edge_agg_lat<<<gE_w, TB, 0, stream>>>(mu_out, lv_out, Xmu, Xlv, row, col, ew, dinv, NE);
}